// PtychoAD_v2_7017976561690
// MI455X (gfx1250) — compile-verified
//
#include <hip/hip_runtime.h>
#include <math.h>

// ---------------------------------------------------------------------------
// Ptychography forward model on MI455X (gfx1250).
// FFTs as complex GEMMs vs the 256x256 DFT matrix using V_WMMA_F32_16X16X4_F32
// (wave32). v2: B-operand K-slabs staged into LDS with
// GLOBAL_LOAD_ASYNC_TO_LDS_B128 (ASYNCcnt) and read back via ds_load; each
// wave computes a 16x32 tile (8 WMMA chains) for higher wmma density.
// Channel chunks of 16 keep ping-pong planes (24MB) resident in the 192MB L2.
// ---------------------------------------------------------------------------

typedef __attribute__((ext_vector_type(2))) float v2f;
typedef __attribute__((ext_vector_type(8))) float v8f;

#define ROI    256
#define PIX    65536        // 256*256
#define NSL    8
#define CH     16           // channels per chunk
#define NCHAN  128          // 32 batch * 4 probe modes
#define KC     32           // K-chunk staged in LDS per iteration

__device__ __forceinline__ v8f wmma_f32(v2f a, v2f b, v8f c) {
  // 8 args: (neg_a, A, neg_b, B, c_mod, C, reuse_a, reuse_b)
  return __builtin_amdgcn_wmma_f32_16x16x4_f32(false, a, false, b,
                                               (short)0, c, false, false);
}

// --------------------------- DFT matrix init -------------------------------
__global__ __launch_bounds__(256) void init_dft(float* __restrict__ Fre,
                                                float* __restrict__ Fim,
                                                float* __restrict__ Gre,
                                                float* __restrict__ Gim) {
  int i = blockIdx.x * 256 + threadIdx.x;     // 0..65535
  int j = i >> 8, k = i & 255;
  int t = (j * k) & 255;                      // angle mod 2*pi
  float ang = (float)t * 0.02454369260617026f; // 2*pi/256
  float s, c;
  sincosf(ang, &s, &c);
  Fre[i] = c;               // exp(-i*ang)
  Fim[i] = -s;
  Gre[i] = c * (1.0f / 256.0f);   // exp(+i*ang)/256  (ifft factor per pass)
  Gim[i] = s * (1.0f / 256.0f);
}

__global__ __launch_bounds__(256) void zero_out(float* __restrict__ o, int n) {
  int i = blockIdx.x * 256 + threadIdx.x;
  if (i < n) o[i] = 0.0f;
}

// ------------------ psi = translated_probe * obj_slice0 --------------------
__global__ __launch_bounds__(256) void compose(const float* __restrict__ probe,
                                               const float* __restrict__ obj,
                                               const float* __restrict__ shifts,
                                               const int* __restrict__ crop,
                                               const int* __restrict__ idxs,
                                               float* __restrict__ Pre,
                                               float* __restrict__ Pim,
                                               int chanBase) {
  int gid = blockIdx.x * 256 + threadIdx.x;   // CH * PIX threads
  int cl  = gid >> 16;                        // local channel
  int pix = gid & 65535;
  int y = pix >> 8, x = pix & 255;
  int g = chanBase + cl;
  int b = g >> 2, m = g & 3;
  int id = idxs[b];
  float tH = shifts[id * 2 + 0];
  float tW = shifts[id * 2 + 1];

  float ys = (float)y - tH, xs = (float)x - tW;
  float y0f = floorf(ys), x0f = floorf(xs);
  int   y0 = (int)y0f,   x0 = (int)x0f;
  float wy = ys - y0f,   wx = xs - x0f;
  float w00 = (1.0f - wy) * (1.0f - wx);
  float w01 = (1.0f - wy) * wx;
  float w10 = wy * (1.0f - wx);
  float w11 = wy * wx;

  auto samp = [&](int yi, int xi, int c) -> float {
    if (yi < 0 || yi >= ROI || xi < 0 || xi >= ROI) return 0.0f;
    return probe[(((m * ROI) + yi) * ROI + xi) * 2 + c];
  };
  float amp = w00 * samp(y0, x0, 0)     + w01 * samp(y0, x0 + 1, 0)
            + w10 * samp(y0 + 1, x0, 0) + w11 * samp(y0 + 1, x0 + 1, 0);
  float ph  = w00 * samp(y0, x0, 1)     + w01 * samp(y0, x0 + 1, 1)
            + w10 * samp(y0 + 1, x0, 1) + w11 * samp(y0 + 1, x0 + 1, 1);
  float sp, cp;
  sincosf(ph, &sp, &cp);
  float pr = amp * cp, pi_ = amp * sp;

  int py = crop[id * 2 + 0], px = crop[id * 2 + 1];
  int ob = ((py + y) * 1024 + (px + x)) * 2;   // slice 0, obj-mode 0
  float oa = obj[ob], op = obj[ob + 1];
  float so, co;
  sincosf(op, &so, &co);
  float orr = oa * co, oii = oa * so;

  Pre[cl * PIX + pix] = pr * orr - pi_ * oii;
  Pim[cl * PIX + pix] = pr * oii + pi_ * orr;
}

// ------------------------ complex GEMM (WMMA fp32) -------------------------
// C = A x B (256x256x256 complex, per channel). aStride/bStride = PIX for
// per-channel operands, 0 for the shared DFT matrices.
// Block = 8 waves = one (channel, 16-row block) of output; wave = 16x32 tile.
// B K-slab (KC x 256, re+im = 64KB) staged in LDS by async copies.
enum { EP_NONE = 0, EP_MULH = 1, EP_MULOBJ = 2, EP_INTENS = 3 };

template <int EP>
__global__ __launch_bounds__(256) void cgemm(
    const float* __restrict__ Ar, const float* __restrict__ Ai, int aStride,
    const float* __restrict__ Br, const float* __restrict__ Bi, int bStride,
    float* __restrict__ Cr, float* __restrict__ Ci,
    const float* __restrict__ Hre, const float* __restrict__ Him,
    const float* __restrict__ obj, int zslice,
    const int* __restrict__ crop, const int* __restrict__ idxs,
    float* __restrict__ outp, int chanBase) {
  __shared__ float sB[2 * KC * 256];          // [0..KC*256) = re, rest = im

  int tid  = threadIdx.x;
  int wave = tid >> 5;
  int lane = tid & 31;
  int cl   = blockIdx.x >> 4;                 // local channel (0..CH-1)
  int tm   = blockIdx.x & 15;                 // 16-row block of output

  const float* ArC = Ar + cl * aStride;
  const float* AiC = Ai + cl * aStride;
  const float* BrC = Br + cl * bStride;
  const float* BiC = Bi + cl * bStride;

  int half = lane >> 4;                       // 0 or 1
  int l15  = lane & 15;
  int arow = tm * 16 + l15;                   // A-fragment row (M)
  int n0   = wave * 32;                       // this wave's N range
  int col0 = n0 + l15;
  int col1 = n0 + 16 + l15;

  v8f c0rr = {}, c0ii = {}, c0ri = {}, c0ir = {};
  v8f c1rr = {}, c1ii = {}, c1ri = {}, c1ir = {};

  const float* aR0 = ArC + arow * 256;
  const float* aI0 = AiC + arow * 256;
  unsigned ldsR = (unsigned)(uintptr_t)&sB[0];
  unsigned ldsI = (unsigned)(uintptr_t)&sB[KC * 256];

  for (int k0 = 0; k0 < 256; k0 += KC) {
    __syncthreads();   // previous chunk's LDS reads complete
    // Stage B K-slab: rows k0..k0+KC-1 are contiguous 32KB per plane.
    unsigned long long gR = (unsigned long long)(const char*)(BrC + k0 * 256);
    unsigned long long gI = (unsigned long long)(const char*)(BiC + k0 * 256);
#pragma unroll
    for (int i = 0; i < 8; ++i) {
      unsigned boff = (unsigned)(tid * 16 + i * 4096);
      asm volatile("global_load_async_to_lds_b128 %0, %1, off"
                   :: "v"(ldsR + boff), "v"(gR + (unsigned long long)boff)
                   : "memory");
      asm volatile("global_load_async_to_lds_b128 %0, %1, off"
                   :: "v"(ldsI + boff), "v"(gI + (unsigned long long)boff)
                   : "memory");
    }
    asm volatile("s_wait_asynccnt 0" ::: "memory");
    __syncthreads();   // slab visible to all waves

#pragma unroll
    for (int kk = 0; kk < KC; kk += 4) {
      int ka = kk + 2 * half;                 // this lane's K pair
      v2f aR = *(const v2f*)(aR0 + k0 + ka);  // contiguous, 8B aligned
      v2f aI = *(const v2f*)(aI0 + k0 + ka);
      int rb = ka * 256;
      v2f bR0, bI0, bR1, bI1;
      bR0.x = sB[rb + col0];            bR0.y = sB[rb + 256 + col0];
      bR1.x = sB[rb + col1];            bR1.y = sB[rb + 256 + col1];
      bI0.x = sB[KC * 256 + rb + col0]; bI0.y = sB[KC * 256 + rb + 256 + col0];
      bI1.x = sB[KC * 256 + rb + col1]; bI1.y = sB[KC * 256 + rb + 256 + col1];
      c0rr = wmma_f32(aR, bR0, c0rr);
      c0ii = wmma_f32(aI, bI0, c0ii);
      c0ri = wmma_f32(aR, bI0, c0ri);
      c0ir = wmma_f32(aI, bR0, c0ir);
      c1rr = wmma_f32(aR, bR1, c1rr);
      c1ii = wmma_f32(aI, bI1, c1ii);
      c1ri = wmma_f32(aR, bI1, c1ri);
      c1ir = wmma_f32(aI, bR1, c1ir);
    }
  }

  int g = chanBase + cl;
  int b = g >> 2;
  int pyv = 0, pxv = 0;
  if constexpr (EP == EP_MULOBJ) {
    int id = idxs[b];
    pyv = crop[id * 2 + 0];
    pxv = crop[id * 2 + 1];
  }

#pragma unroll
  for (int s = 0; s < 2; ++s) {
    int colg = n0 + s * 16 + l15;
#pragma unroll
    for (int r = 0; r < 8; ++r) {
      int rowg = tm * 16 + r + 8 * half;      // C/D layout: halves split M
      float vr, vi;
      if (s == 0) { vr = c0rr[r] - c0ii[r];  vi = c0ri[r] + c0ir[r]; }
      else        { vr = c1rr[r] - c1ii[r];  vi = c1ri[r] + c1ir[r]; }
      if constexpr (EP == EP_MULH) {
        float hr = Hre[rowg * 256 + colg];
        float hm = Him[rowg * 256 + colg];
        float nr = vr * hr - vi * hm;
        float ni = vr * hm + vi * hr;
        vr = nr; vi = ni;
      } else if constexpr (EP == EP_MULOBJ) {
        int ob = ((zslice * 1024 + (pyv + rowg)) * 1024 + (pxv + colg)) * 2;
        float oa = obj[ob], op = obj[ob + 1];
        float so, co;
        sincosf(op, &so, &co);
        float orr2 = oa * co, oii2 = oa * so;
        float nr = vr * orr2 - vi * oii2;
        float ni = vr * oii2 + vi * orr2;
        vr = nr; vi = ni;
      }
      if constexpr (EP == EP_INTENS) {
        atomicAdd(outp + b * PIX + rowg * 256 + colg, vr * vr + vi * vi);
      } else {
        Cr[cl * PIX + rowg * 256 + colg] = vr;
        Ci[cl * PIX + rowg * 256 + colg] = vi;
      }
    }
  }
}

// ------------------------------ launcher -----------------------------------
extern "C" void kernel_launch(void* const* d_in, const int* in_sizes, int n_in,
                              void* d_out, int out_size, void* d_ws,
                              size_t ws_size, hipStream_t stream) {
  const float* obj    = (const float*)d_in[0];  // (1,8,1024,1024,2)
  const float* probe  = (const float*)d_in[1];  // (4,256,256,2)
  const float* shifts = (const float*)d_in[2];  // (1024,2)
  const int*   crop   = (const int*)  d_in[3];  // (1024,2)
  const float* Hre    = (const float*)d_in[4];  // (256,256)
  const float* Him    = (const float*)d_in[5];  // (256,256)
  const int*   idxs   = (const int*)  d_in[6];  // (32,)
  float* outp = (float*)d_out;                  // (32,256,256)

  float* ws  = (float*)d_ws;
  float* Fre = ws; ws += PIX;
  float* Fim = ws; ws += PIX;
  float* Gre = ws; ws += PIX;
  float* Gim = ws; ws += PIX;
  float* Pre = ws; ws += CH * PIX;
  float* Pim = ws; ws += CH * PIX;
  float* Qre = ws; ws += CH * PIX;
  float* Qim = ws; ws += CH * PIX;
  float* Rre = ws; ws += CH * PIX;
  float* Rim = ws; ws += CH * PIX;

  init_dft<<<PIX / 256, 256, 0, stream>>>(Fre, Fim, Gre, Gim);
  zero_out<<<(32 * PIX) / 256, 256, 0, stream>>>(outp, 32 * PIX);

  const int gemmBlocks = CH * 16;  // one block per (channel, 16-row block)

  for (int chunk = 0; chunk < NCHAN / CH; ++chunk) {
    int cb = chunk * CH;
    compose<<<CH * PIX / 256, 256, 0, stream>>>(probe, obj, shifts, crop, idxs,
                                                Pre, Pim, cb);
    for (int z = 1; z < NSL; ++z) {
      // T = F * psi
      cgemm<EP_NONE><<<gemmBlocks, 256, 0, stream>>>(
          Fre, Fim, 0, Pre, Pim, PIX, Qre, Qim,
          nullptr, nullptr, nullptr, 0, nullptr, nullptr, nullptr, cb);
      // U = (T * F) .* H
      cgemm<EP_MULH><<<gemmBlocks, 256, 0, stream>>>(
          Qre, Qim, PIX, Fre, Fim, 0, Rre, Rim,
          Hre, Him, nullptr, 0, nullptr, nullptr, nullptr, cb);
      // V = Finv * U
      cgemm<EP_NONE><<<gemmBlocks, 256, 0, stream>>>(
          Gre, Gim, 0, Rre, Rim, PIX, Qre, Qim,
          nullptr, nullptr, nullptr, 0, nullptr, nullptr, nullptr, cb);
      // psi = (V * Finv) .* obj_z
      cgemm<EP_MULOBJ><<<gemmBlocks, 256, 0, stream>>>(
          Qre, Qim, PIX, Gre, Gim, 0, Pre, Pim,
          nullptr, nullptr, obj, z, crop, idxs, nullptr, cb);
    }
    // T = F * psi
    cgemm<EP_NONE><<<gemmBlocks, 256, 0, stream>>>(
        Fre, Fim, 0, Pre, Pim, PIX, Qre, Qim,
        nullptr, nullptr, nullptr, 0, nullptr, nullptr, nullptr, cb);
    // out[b] += |T * F|^2
    cgemm<EP_INTENS><<<gemmBlocks, 256, 0, stream>>>(
        Qre, Qim, PIX, Fre, Fim, 0, Rre, Rim,
        nullptr, nullptr, nullptr, 0, nullptr, nullptr, outp, cb);
  }
}